// FeaST1_50371376447896
// MI455X (gfx1250) — compile-verified
//
#include <hip/hip_runtime.h>
#include <hip/hip_bf16.h>

typedef __attribute__((ext_vector_type(2))) float v2f;
typedef __attribute__((ext_vector_type(8))) float v8f;

#define IN_CH  128
#define OUT_CH 128

// ---------------------------------------------------------------------------
// Kernel 0: zero the accumulator workspace (agg + cnt) every call.
// ---------------------------------------------------------------------------
__global__ void feast_zero_kernel(float* __restrict__ agg, float* __restrict__ cnt,
                                  int nAgg, int nCnt) {
    int gid = blockIdx.x * blockDim.x + threadIdx.x;
    if (gid < nAgg) agg[gid] = 0.0f;
    if (gid < nCnt) cnt[gid] = 0.0f;
}

// ---------------------------------------------------------------------------
// Kernel 1: Y = X @ W  via V_WMMA_F32_16X16X4_F32 (fp32 matrix pipe).
// Block = 256 threads = 8 wave32s. Block computes a 16-row x 128-col slab:
// the 16x128 X tile is staged in LDS once, each wave owns one 16x16 N-tile
// and iterates K = 0..128 in steps of 4.
// ---------------------------------------------------------------------------
__global__ void feast_gemm_kernel(const float* __restrict__ X,
                                  const float* __restrict__ W,
                                  float* __restrict__ Y, int N) {
    __shared__ float xs[16][IN_CH + 4];   // +4 pad to spread LDS banks

    const int tid  = threadIdx.x;
    const int wave = tid >> 5;            // 0..7  -> N-tile
    const int lane = tid & 31;
    const int l16  = lane & 15;
    const int half = lane >> 4;           // 0: lanes 0-15, 1: lanes 16-31

    const int row0 = blockIdx.x * 16;

    // Stage 16x128 tile of X into LDS (2048 floats, 8 per thread).
    for (int i = tid; i < 16 * IN_CH; i += 256) {
        int r = i >> 7;          // /128
        int c = i & 127;
        int gr = row0 + r;
        xs[r][c] = (gr < N) ? X[(size_t)gr * IN_CH + c] : 0.0f;
    }
    __syncthreads();

    const int n0 = wave * 16;             // this wave's output columns
    v8f acc = {};

    for (int k = 0; k < IN_CH; k += 4) {
        // A fragment: 16x4.  VGPR0: K = k+2*half, VGPR1: K = k+2*half+1 (ISA 7.12.2)
        v2f a, b;
        a.x = xs[l16][k + 2 * half];
        a.y = xs[l16][k + 2 * half + 1];
        // B fragment: 4x16 of W, rows striped across lanes.
        b.x = W[(size_t)(k + 2 * half)     * OUT_CH + n0 + l16];
        b.y = W[(size_t)(k + 2 * half + 1) * OUT_CH + n0 + l16];
        acc = __builtin_amdgcn_wmma_f32_16x16x4_f32(
                  /*neg_a=*/false, a, /*neg_b=*/false, b,
                  /*c_mod=*/(short)0, acc, /*reuse_a=*/false, /*reuse_b=*/false);
    }

    // D layout: VGPR r holds rows (r, r+8); lanes 0-15 -> M=r, 16-31 -> M=r+8.
    const int col = n0 + l16;
#pragma unroll
    for (int r = 0; r < 8; ++r) {
        int row = row0 + r + 8 * half;
        if (row < N) Y[(size_t)row * OUT_CH + col] = acc[r];
    }
}

// ---------------------------------------------------------------------------
// Kernel 2: edge scatter.  One wave32 per edge: each lane atomically adds
// 4 channels of Y[src] into agg[dst]; lane 0 bumps the degree counter.
// ---------------------------------------------------------------------------
__global__ void feast_scatter_kernel(const int* __restrict__ ei,
                                     const float* __restrict__ Y,
                                     float* __restrict__ agg,
                                     float* __restrict__ cnt, int E) {
    const long long gid  = (long long)blockIdx.x * blockDim.x + threadIdx.x;
    const int       edge = (int)(gid >> 5);
    const int       lane = (int)(gid & 31);
    if (edge >= E) return;

    const int src = ei[edge];        // edge_index[0][e]
    const int dst = ei[E + edge];    // edge_index[1][e]

    // Pull a future edge's source row toward this WGP (global_prefetch_b8).
    int pe = edge + 512;
    if (pe < E) {
        int ps = ei[pe];
        __builtin_prefetch(&Y[(size_t)ps * OUT_CH + lane * 4], 0, 1);
    }

    const float* ys = Y   + (size_t)src * OUT_CH;
    float*       ad = agg + (size_t)dst * OUT_CH;
#pragma unroll
    for (int j = 0; j < 4; ++j) {
        int c = lane + 32 * j;
        atomicAdd(&ad[c], ys[c]);
    }
    if (lane == 0) atomicAdd(&cnt[dst], 1.0f);
}

// ---------------------------------------------------------------------------
// Kernel 3: finalize.  Fold in self-loop (Y[i], +1 to degree), mean, bias, relu.
// cnt+1 >= 1 so the max(cnt,1) clamp of the reference is automatic.
// ---------------------------------------------------------------------------
__global__ void feast_finalize_kernel(const float* __restrict__ Y,
                                      const float* __restrict__ agg,
                                      const float* __restrict__ cnt,
                                      const float* __restrict__ bias,
                                      float* __restrict__ out, int N) {
    int gid = blockIdx.x * blockDim.x + threadIdx.x;
    if (gid >= N * OUT_CH) return;
    int node = gid >> 7;       // /128
    int f    = gid & 127;
    float deg = cnt[node] + 1.0f;
    float v   = (agg[gid] + Y[gid]) / deg + bias[f];
    out[gid]  = fmaxf(v, 0.0f);
}

// ---------------------------------------------------------------------------
extern "C" void kernel_launch(void* const* d_in, const int* in_sizes, int n_in,
                              void* d_out, int out_size, void* d_ws, size_t ws_size,
                              hipStream_t stream) {
    const float* x      = (const float*)d_in[0];
    const int*   ei     = (const int*)d_in[1];
    const float* weight = (const float*)d_in[2];
    // d_in[3] = u, d_in[4] = c : mathematically irrelevant for HEADS==1
    const float* bias   = (const float*)d_in[5];
    float*       out    = (float*)d_out;

    const int N = in_sizes[0] / IN_CH;   // 50000
    const int E = in_sizes[1] / 2;       // 600000

    // Workspace layout: Y | agg | cnt
    float* Y   = (float*)d_ws;
    float* agg = Y   + (size_t)N * OUT_CH;
    float* cnt = agg + (size_t)N * OUT_CH;

    const int nAgg = N * OUT_CH;

    // 0) zero accumulators (deterministic per call)
    {
        int threads = 256;
        int blocks  = (nAgg + threads - 1) / threads;
        feast_zero_kernel<<<blocks, threads, 0, stream>>>(agg, cnt, nAgg, N);
    }
    // 1) Y = X @ W  (WMMA f32)
    {
        int blocks = (N + 15) / 16;      // 3125
        feast_gemm_kernel<<<blocks, 256, 0, stream>>>(x, weight, Y, N);
    }
    // 2) edge scatter (wave per edge)
    {
        long long threads_total = (long long)E * 32;
        int threads = 256;
        int blocks  = (int)((threads_total + threads - 1) / threads);
        feast_scatter_kernel<<<blocks, threads, 0, stream>>>(ei, Y, agg, cnt, E);
    }
    // 3) finalize: mean + self-loop + bias + relu
    {
        int threads = 256;
        int blocks  = (nAgg + threads - 1) / threads;
        feast_finalize_kernel<<<blocks, threads, 0, stream>>>(Y, agg, cnt, bias, out, N);
    }
}